// BLNN_21199958573785
// MI455X (gfx1250) — compile-verified
//
#include <hip/hip_runtime.h>

typedef float v2f __attribute__((ext_vector_type(2)));
typedef float v8f __attribute__((ext_vector_type(8)));

#define B_     256
#define H_     64
#define IN_    16
#define CHUNK_ 128
#define MAXIT_ 1000

// LDS strides (padded to dodge 64-bank conflicts on column-ish accesses)
#define SX_S 17   // x, z, resid rows of 16
#define SP_S 65   // h2/u, pre1/a rows of 64
#define SW_S 65   // Wz1p 64x64
#define SY_S 17   // Wy0, Wy1 rows of 16

enum {
  OFF_X    = 0,
  OFF_Z    = OFF_X    + B_ * SX_S,
  OFF_P0   = OFF_Z    + B_ * SX_S,      // h2 = softplus(pre0), later u       (chunk-local)
  OFF_P1   = OFF_P0   + CHUNK_ * SP_S,  // pre1, later a = Wz2p*sigmoid(pre1) (chunk-local)
  OFF_R    = OFF_P1   + CHUNK_ * SP_S,  // residual chunk
  OFF_WZ1  = OFF_R    + CHUNK_ * SX_S,  // clipped Wz1  64x64
  OFF_WY0  = OFF_WZ1  + H_ * SW_S,
  OFF_WY1  = OFF_WY0  + H_ * SY_S,
  OFF_WY2  = OFF_WY1  + H_ * SY_S,      // 16
  OFF_WZ2  = OFF_WY2  + IN_,            // clipped, 64
  OFF_B0   = OFF_WZ2  + H_,
  OFF_B1   = OFF_B0   + H_,
  OFF_B2   = OFF_B1   + H_,
  OFF_S2   = OFF_B2   + 1,              // sigmoid(pre2), chunk-local 128
  OFF_NRM  = OFF_S2   + CHUNK_,         // per-row residual norms, 256
  OFF_PART = OFF_NRM  + B_,             // pre2 partials / reduce partials, 256
  OFF_FLG  = OFF_PART + B_,
  SMEM_FLOATS = OFF_FLG + 1             // ~135.6 KB
};

__device__ __forceinline__ float softplus_(float x) {
  // matches jax.nn.softplus numerics: max(x,0) + log1p(exp(-|x|))
  return fmaxf(x, 0.0f) + log1pf(expf(-fabsf(x)));
}
__device__ __forceinline__ float sigmoid_(float x) {
  return 1.0f / (1.0f + expf(-x));
}
// sigmoid(pre0) recovered exactly from h2 = softplus(pre0):
//   e^{-softplus(x)} = 1/(1+e^x) = 1 - sigmoid(x)
__device__ __forceinline__ float sig_from_sp_(float h2) {
  return 1.0f - expf(-h2);
}

__device__ __forceinline__ v8f wmma4(v2f a, v2f b, v8f c) {
  // V_WMMA_F32_16X16X4_F32 : D = A(16x4,f32) * B(4x16,f32) + C(16x16,f32)
  return __builtin_amdgcn_wmma_f32_16x16x4_f32(false, a, false, b, (short)0, c,
                                               false, false);
}

__global__ __launch_bounds__(256) void BLNN_21199958573785_kernel(
    const float* __restrict__ z,
    const float* __restrict__ wy0w, const float* __restrict__ wy0b,
    const float* __restrict__ wy1w, const float* __restrict__ wy1b,
    const float* __restrict__ wz1w,
    const float* __restrict__ wy2w, const float* __restrict__ wy2b,
    const float* __restrict__ wz2w,
    float* __restrict__ out) {
  extern __shared__ float smem[];
  const int tid  = threadIdx.x;
  const int lane = tid & 31;
  const int wv   = tid >> 5;      // wave id 0..7
  const int half = lane >> 4;     // 0: lanes 0-15, 1: lanes 16-31
  const int l16  = lane & 15;
  const int ni0  = wv & 3;        // this wave's column tile (fixed -> B-frag reuse)
  const int mi0  = wv >> 2;       // row-tile base

  // ---------------- load weights / inputs into LDS; x0 = ones ----------------
  for (int idx = tid; idx < B_ * IN_; idx += 256) {
    int r = idx >> 4, c = idx & 15;
    smem[OFF_X + r * SX_S + c] = 1.0f;
    smem[OFF_Z + r * SX_S + c] = z[idx];
  }
  for (int idx = tid; idx < H_ * IN_; idx += 256) {
    int r = idx >> 4, c = idx & 15;
    smem[OFF_WY0 + r * SY_S + c] = wy0w[idx];
    smem[OFF_WY1 + r * SY_S + c] = wy1w[idx];
  }
  for (int idx = tid; idx < H_ * H_; idx += 256) {
    int r = idx >> 6, c = idx & 63;
    smem[OFF_WZ1 + r * SW_S + c] = fmaxf(wz1w[idx], 0.0f);  // PositiveLinear clamp
  }
  if (tid < H_) {
    smem[OFF_B0 + tid]  = wy0b[tid];
    smem[OFF_B1 + tid]  = wy1b[tid];
    smem[OFF_WZ2 + tid] = fmaxf(wz2w[tid], 0.0f);
  }
  if (tid < IN_) smem[OFF_WY2 + tid] = wy2w[tid];
  if (tid == 0) { smem[OFF_B2] = wy2b[0]; smem[OFF_FLG] = 0.0f; }
  __syncthreads();

  for (int it = 0; it < MAXIT_; ++it) {
    for (int ch = 0; ch < 2; ++ch) {
      const int rowbase = ch * CHUNK_;

      // ---- Phase 1: h2 = softplus(x @ Wy0^T + b0)   (M=128, N=64, K=16) ----
      // wave handles 4 row-tiles in one fixed column tile -> B frags stay in VGPRs
#pragma unroll
      for (int j = 0; j < 4; ++j) {
        const int lm0 = (mi0 + 2 * j) * 16, n0 = ni0 * 16, gm0 = rowbase + lm0;
        v8f acc;
        { float bv = smem[OFF_B0 + n0 + l16];
#pragma unroll
          for (int r = 0; r < 8; ++r) acc[r] = bv; }
#pragma unroll
        for (int kk = 0; kk < IN_; kk += 4) {
          const float* pa = &smem[OFF_X + (gm0 + l16) * SX_S + kk + 2 * half];
          v2f a; a.x = pa[0]; a.y = pa[1];
          // B[k=i][n=h] = Wy0[h][i]  (transposed access)
          const float* pb = &smem[OFF_WY0 + (n0 + l16) * SY_S + kk + 2 * half];
          v2f b; b.x = pb[0]; b.y = pb[1];
          acc = wmma4(a, b, acc);
        }
#pragma unroll
        for (int r = 0; r < 8; ++r)
          smem[OFF_P0 + (lm0 + r + 8 * half) * SP_S + n0 + l16] = softplus_(acc[r]);
      }
      __syncthreads();

      // ---- Phase 2: pre1 = h2 @ Wz1p^T + x @ Wy1^T + b1 ----
#pragma unroll
      for (int j = 0; j < 4; ++j) {
        const int lm0 = (mi0 + 2 * j) * 16, n0 = ni0 * 16, gm0 = rowbase + lm0;
        v8f acc;
        { float bv = smem[OFF_B1 + n0 + l16];
#pragma unroll
          for (int r = 0; r < 8; ++r) acc[r] = bv; }
#pragma unroll
        for (int kk = 0; kk < H_; kk += 4) {   // K over h: A = h2 (plain load now)
          const float* pa = &smem[OFF_P0 + (lm0 + l16) * SP_S + kk + 2 * half];
          v2f a; a.x = pa[0]; a.y = pa[1];
          // B[k=h][n=o] = Wz1p[o][h] (transposed access)
          const float* pb = &smem[OFF_WZ1 + (n0 + l16) * SW_S + kk + 2 * half];
          v2f b; b.x = pb[0]; b.y = pb[1];
          acc = wmma4(a, b, acc);
        }
#pragma unroll
        for (int kk = 0; kk < IN_; kk += 4) {  // K over i: A = x
          const float* pa = &smem[OFF_X + (gm0 + l16) * SX_S + kk + 2 * half];
          v2f a; a.x = pa[0]; a.y = pa[1];
          const float* pb = &smem[OFF_WY1 + (n0 + l16) * SY_S + kk + 2 * half];
          v2f b; b.x = pb[0]; b.y = pb[1];
          acc = wmma4(a, b, acc);
        }
#pragma unroll
        for (int r = 0; r < 8; ++r)
          smem[OFF_P1 + (lm0 + r + 8 * half) * SP_S + n0 + l16] = acc[r];
      }
      __syncthreads();

      // ---- Phase 3/4 (2 threads per row): pre2 partials; a = Wz2p*sigmoid(pre1) ----
      {
        const int r   = tid & (CHUNK_ - 1);
        const int seg = tid >> 7;             // h-half 0/1
        const int gr  = rowbase + r;
        const int h0  = seg * 32;
        float part = 0.0f;
#pragma unroll 8
        for (int h = h0; h < h0 + 32; ++h) {
          float p  = smem[OFF_P1 + r * SP_S + h];
          float wz = smem[OFF_WZ2 + h];
          part += softplus_(p) * wz;
          smem[OFF_P1 + r * SP_S + h] = wz * sigmoid_(p);   // overwrite pre1 with a
        }
        const int i0 = seg * 8;
#pragma unroll
        for (int i = i0; i < i0 + 8; ++i)
          part += smem[OFF_X + gr * SX_S + i] * smem[OFF_WY2 + i];
        smem[OFF_PART + tid] = part;
      }
      __syncthreads();

      // ---- s2 combine (fused here; touches only PART/S2, disjoint from GEMM) ----
      if (tid < CHUNK_) {
        float acc2 = smem[OFF_PART + tid] + smem[OFF_PART + tid + CHUNK_] + smem[OFF_B2];
        smem[OFF_S2 + tid] = sigmoid_(acc2);
      }

      // ---- Phase 5: c = a @ Wz1p (M=128,N=64,K=64); fuse u = c * sigmoid(pre0) ----
#pragma unroll
      for (int j = 0; j < 4; ++j) {
        const int lm0 = (mi0 + 2 * j) * 16, n0 = ni0 * 16;
        v8f acc;
#pragma unroll
        for (int r = 0; r < 8; ++r) acc[r] = 0.0f;
#pragma unroll
        for (int kk = 0; kk < H_; kk += 4) {
          const float* pa = &smem[OFF_P1 + (lm0 + l16) * SP_S + kk + 2 * half];
          v2f a; a.x = pa[0]; a.y = pa[1];
          // B[k=o][n=h] = Wz1p[o][h] (natural row-major)
          const float* pb = &smem[OFF_WZ1 + (kk + 2 * half) * SW_S + n0 + l16];
          v2f b; b.x = pb[0]; b.y = pb[SW_S];
          acc = wmma4(a, b, acc);
        }
#pragma unroll
        for (int r = 0; r < 8; ++r) {
          const int lr = lm0 + r + 8 * half;
          float h2v = smem[OFF_P0 + lr * SP_S + n0 + l16];
          smem[OFF_P0 + lr * SP_S + n0 + l16] = acc[r] * sig_from_sp_(h2v);
        }
      }
      __syncthreads();

      // ---- Phase 6: term = u@Wy0 + a@Wy1 (M=128,N=16,K=64+64); fuse resid ----
      {
        const int lm0 = wv * 16;   // one 16x16 output tile per wave
        v8f acc;
#pragma unroll
        for (int r = 0; r < 8; ++r) acc[r] = 0.0f;
#pragma unroll
        for (int kk = 0; kk < H_; kk += 4) {   // u @ Wy0
          const float* pa = &smem[OFF_P0 + (lm0 + l16) * SP_S + kk + 2 * half];
          v2f a; a.x = pa[0]; a.y = pa[1];
          const float* pb = &smem[OFF_WY0 + (kk + 2 * half) * SY_S + l16];
          v2f b; b.x = pb[0]; b.y = pb[SY_S];
          acc = wmma4(a, b, acc);
        }
#pragma unroll
        for (int kk = 0; kk < H_; kk += 4) {   // a @ Wy1
          const float* pa = &smem[OFF_P1 + (lm0 + l16) * SP_S + kk + 2 * half];
          v2f a; a.x = pa[0]; a.y = pa[1];
          const float* pb = &smem[OFF_WY1 + (kk + 2 * half) * SY_S + l16];
          v2f b; b.x = pb[0]; b.y = pb[SY_S];
          acc = wmma4(a, b, acc);
        }
#pragma unroll
        for (int r = 0; r < 8; ++r) {
          const int lr = lm0 + r + 8 * half;
          const int gr = rowbase + lr;
          float g  = smem[OFF_S2 + lr] * (acc[r] + smem[OFF_WY2 + l16]);
          float f1 = g + smem[OFF_X + gr * SX_S + l16] * (1.0f / 1.5f);
          smem[OFF_R + lr * SX_S + l16] = smem[OFF_Z + gr * SX_S + l16] - f1;
        }
      }
      __syncthreads();

      // ---- Phase 7 (per row): row norm; x += (3/(it+1)) * resid ----
      if (tid < CHUNK_) {
        const int gr = rowbase + tid;
        const float stepv = 3.0f / ((float)it + 1.0f);
        float ss = 0.0f;
#pragma unroll
        for (int i = 0; i < IN_; ++i) {
          float rv = smem[OFF_R + tid * SX_S + i];
          ss += rv * rv;
          smem[OFF_X + gr * SX_S + i] += stepv * rv;
        }
        smem[OFF_NRM + gr] = sqrtf(ss);
      }
      __syncthreads();
    }  // chunks

    // ---- convergence: mean row norm < 1e-3 (wave32 shfl tree, deterministic) ----
    {
      float v = smem[OFF_NRM + tid];
#pragma unroll
      for (int off = 16; off > 0; off >>= 1) v += __shfl_xor(v, off, 32);
      if (lane == 0) smem[OFF_PART + wv] = v;
    }
    __syncthreads();
    if (tid == 0) {
      float s = 0.0f;
#pragma unroll
      for (int w = 0; w < 8; ++w) s += smem[OFF_PART + w];
      smem[OFF_FLG] = (s * (1.0f / 256.0f) < 1e-3f) ? 1.0f : 0.0f;
    }
    __syncthreads();
    if (smem[OFF_FLG] != 0.0f) break;   // x frozen once converged == reference scan
  }

  // ---- out = x + 0.5 * z ----
  for (int idx = tid; idx < B_ * IN_; idx += 256) {
    int r = idx >> 4, c = idx & 15;
    out[idx] = smem[OFF_X + r * SX_S + c] + 0.5f * smem[OFF_Z + r * SX_S + c];
  }
}

extern "C" void kernel_launch(void* const* d_in, const int* in_sizes, int n_in,
                              void* d_out, int out_size, void* d_ws, size_t ws_size,
                              hipStream_t stream) {
  (void)in_sizes; (void)n_in; (void)out_size; (void)d_ws; (void)ws_size;
  const float* z    = (const float*)d_in[0];
  const float* wy0w = (const float*)d_in[1];
  const float* wy0b = (const float*)d_in[2];
  const float* wy1w = (const float*)d_in[3];
  const float* wy1b = (const float*)d_in[4];
  const float* wz1w = (const float*)d_in[5];
  const float* wy2w = (const float*)d_in[6];
  const float* wy2b = (const float*)d_in[7];
  const float* wz2w = (const float*)d_in[8];
  float* out = (float*)d_out;

  const size_t smem_bytes = (size_t)SMEM_FLOATS * sizeof(float);
  // Single persistent workgroup: 8 wave32s on one WGP; whole solve in LDS.
  BLNN_21199958573785_kernel<<<1, 256, smem_bytes, stream>>>(
      z, wy0w, wy0b, wy1w, wy1b, wz1w, wy2w, wy2b, wz2w, out);
}